// GroupPoints_65309272703443
// MI455X (gfx1250) — compile-verified
//
#include <hip/hip_runtime.h>
#include <math.h>

// Problem constants from the reference
#define BB   4
#define NS   16384
#define NT   4096
#define DD   3
#define KK   32
#define RAD  0.2f
#define EPSD 1e-9f

#define TGT_PER_BLOCK 128   // 4 waves * 32 targets
#define SRC_CHUNK     256   // sources staged in LDS per iteration
#define SCR_STRIDE    20    // dwords; 80B row stride -> aligned b128 reads, conflict-free

typedef __attribute__((ext_vector_type(2))) float v2f;
typedef __attribute__((ext_vector_type(8))) float v8f;

__device__ __forceinline__ unsigned lds_off(const void* p) {
    // addrspacecast LDS->flat keeps the LDS byte offset in the low 32 bits
    return (unsigned)(uintptr_t)p;
}

__global__ __launch_bounds__(TGT_PER_BLOCK)
void group_points_knn_wmma(const float* __restrict__ source,
                           const float* __restrict__ target,
                           float* __restrict__ outPatches,  // [B,NT,K,3]
                           float* __restrict__ outIdx,      // [B,NT,K,2] (as float)
                           float* __restrict__ outSize,     // [B,NT]
                           float* __restrict__ outRad,      // [B]
                           float* __restrict__ outDist)     // [B,NT,K]
{
    __shared__ float tgt[TGT_PER_BLOCK * 4];                            // x,y,z,|t|^2
    __shared__ __attribute__((aligned(16))) float srcb[SRC_CHUNK * 3];  // raw xyz (async DMA)
    __shared__ __attribute__((aligned(16))) float scr[4][32 * SCR_STRIDE];

    const int blocksPerBatch = NT / TGT_PER_BLOCK;      // 32
    const int b     = blockIdx.x / blocksPerBatch;
    const int tbase = (blockIdx.x % blocksPerBatch) * TGT_PER_BLOCK;

    const int tid  = threadIdx.x;
    const int lane = tid & 31;
    const int wave = tid >> 5;

    // ---- stage this block's 128 targets (coords + |t|^2) into LDS ----
    for (int i = tid; i < TGT_PER_BLOCK; i += blockDim.x) {
        const float* tp = target + ((size_t)b * NT + tbase + i) * DD;
        float x = tp[0], y = tp[1], z = tp[2];
        tgt[i * 4 + 0] = x;
        tgt[i * 4 + 1] = y;
        tgt[i * 4 + 2] = z;
        tgt[i * 4 + 3] = x * x + y * y + z * z;
    }
    __syncthreads();

    // ---- build A fragments for this wave's 32 targets ----
    // A layout (16x4 f32): lanes 0-15 -> {K0,K1}, lanes 16-31 -> {K2,K3}, M = lane&15
    const int  m     = lane & 15;
    const bool hi    = lane >= 16;
    const int  rbase = hi ? 8 : 0;

    const float* t0p = &tgt[(wave * 32 + m) * 4];
    const float* t1p = &tgt[(wave * 32 + 16 + m) * 4];
    v2f a0, a1;
    a0.x = hi ? (-2.0f * t0p[2]) : (-2.0f * t0p[0]);   // K2=-2tz | K0=-2tx
    a0.y = hi ? 1.0f             : (-2.0f * t0p[1]);   // K3=1    | K1=-2ty
    a1.x = hi ? (-2.0f * t1p[2]) : (-2.0f * t1p[0]);
    a1.y = hi ? 1.0f             : (-2.0f * t1p[1]);

    // C init = |t|^2 per row  ->  WMMA emits true squared distances
    v8f cinit0, cinit1;
#pragma unroll
    for (int r = 0; r < 8; ++r) {
        cinit0[r] = tgt[(wave * 32 + rbase + r) * 4 + 3];
        cinit1[r] = tgt[(wave * 32 + 16 + rbase + r) * 4 + 3];
    }

    // ---- per-lane running top-K (sorted ascending) ----
    float best_d[KK];
    int   best_i[KK];
#pragma unroll
    for (int q = 0; q < KK; ++q) { best_d[q] = 3.0e38f; best_i[q] = 0; }

    const unsigned srcbOff = lds_off(&srcb[0]);

    // ---- stream sources through LDS via async DMA, WMMA tiles, insert ----
    for (int s0 = 0; s0 < NS; s0 += SRC_CHUNK) {
        __syncthreads();                      // all waves done reading srcb

        // async global->LDS: 256 points * 12B = 3072B = 192 x b128 per-lane copies
        {
            const uint64_t gbase =
                (uint64_t)(uintptr_t)(source + ((size_t)b * NS + s0) * DD);
            for (int i = tid; i < (SRC_CHUNK * DD * 4) / 16; i += TGT_PER_BLOCK) {
                unsigned la = srcbOff + (unsigned)i * 16u;
                uint64_t ga = gbase + (uint64_t)i * 16u;
                asm volatile("global_load_async_to_lds_b128 %0, %1, off"
                             :: "v"(la), "v"(ga) : "memory");
            }
            asm volatile("s_wait_asynccnt 0" ::: "memory");
        }
        __syncthreads();

        for (int sub = 0; sub < SRC_CHUNK / 16; ++sub) {
            // B layout (4x16 f32): lanes 0-15 -> {K0,K2}, lanes 16-31 -> {K1,K3}, N = lane&15
            const float* sp = &srcb[(sub * 16 + m) * 3];
            float sx = sp[0], sy = sp[1], sz = sp[2];
            v2f bf;
            bf.x = hi ? sy                          : sx;   // K1=sy    | K0=sx
            bf.y = hi ? (sx * sx + sy * sy + sz * sz) : sz; // K3=|s|^2 | K2=sz

            v8f c0 = cinit0, c1 = cinit1;
            c0 = __builtin_amdgcn_wmma_f32_16x16x4_f32(false, a0, false, bf,
                                                       (short)0, c0, false, false);
            c1 = __builtin_amdgcn_wmma_f32_16x16x4_f32(false, a1, false, bf,
                                                       (short)0, c1, false, false);

            // scatter C tiles (row M = rbase+r [+16], col = lane&15) to padded LDS
#pragma unroll
            for (int r = 0; r < 8; ++r) {
                scr[wave][(rbase + r) * SCR_STRIDE + m]      = c0[r];
                scr[wave][(16 + rbase + r) * SCR_STRIDE + m] = c1[r];
            }
            asm volatile("s_wait_dscnt 0" ::: "memory");  // cross-lane LDS RAW

            // lane owns row=lane: fetch its 16 candidates with 4 aligned b128 loads
            const int sbase = s0 + sub * 16;
            const float4* rowp =
                reinterpret_cast<const float4*>(&scr[wave][lane * SCR_STRIDE]);
#pragma unroll
            for (int jj = 0; jj < 4; ++jj) {
                float4 v = rowp[jj];
                float dv[4] = {v.x, v.y, v.z, v.w};
#pragma unroll
                for (int e = 0; e < 4; ++e) {
                    float d = dv[e];
                    if (d < best_d[KK - 1]) {
                        float cd = d;
                        int   ci = sbase + jj * 4 + e;
#pragma unroll
                        for (int q = 0; q < KK; ++q) {
                            bool  sw = cd < best_d[q];
                            float td = best_d[q];
                            int   ti = best_i[q];
                            best_d[q] = sw ? cd : td;
                            best_i[q] = sw ? ci : ti;
                            cd = sw ? td : cd;
                            ci = sw ? ti : ci;
                        }
                    }
                }
            }
        }
    }

    // ---- finalize: radius mask, gather neighbor coords, write outputs ----
    const int    tloc = wave * 32 + lane;           // lane owns row = lane
    const size_t bt   = (size_t)b * NT + tbase + tloc;
    const float  tx = tgt[tloc * 4 + 0];
    const float  ty = tgt[tloc * 4 + 1];
    const float  tz = tgt[tloc * 4 + 2];
    const float  invR = 1.0f / RAD;
    const float  r2   = RAD * RAD;

    float* pP = outPatches + bt * KK * 3;
    float* pI = outIdx     + bt * KK * 2;
    float* pD = outDist    + bt * KK;
    float  cnt = 0.0f;

#pragma unroll 4
    for (int q = 0; q < KK; ++q) {
        float sq  = best_d[q];
        int   id  = best_i[q];
        bool  msk = r2 >= sq;
        const float* sp = source + ((size_t)b * NS + id) * DD;
        float gx = msk ? sp[0] * invR : 0.0f;
        float gy = msk ? sp[1] * invR : 0.0f;
        float gz = msk ? sp[2] * invR : 0.0f;
        pP[q * 3 + 0] = gx - tx * invR;
        pP[q * 3 + 1] = gy - ty * invR;
        pP[q * 3 + 2] = gz - tz * invR;
        pI[q * 2 + 0] = (float)b;
        pI[q * 2 + 1] = msk ? (float)id : -1.0f;
        pD[q] = sqrtf(fmaxf(sq, EPSD)) * invR;
        cnt += msk ? 1.0f : 0.0f;
    }
    outSize[bt] = cnt;

    if (blockIdx.x == 0 && tid < BB) outRad[tid] = RAD;
}

extern "C" void kernel_launch(void* const* d_in, const int* in_sizes, int n_in,
                              void* d_out, int out_size, void* d_ws, size_t ws_size,
                              hipStream_t stream) {
    (void)in_sizes; (void)n_in; (void)d_ws; (void)ws_size; (void)out_size;

    const float* source = (const float*)d_in[0];   // [B,NS,3]
    const float* target = (const float*)d_in[1];   // [B,NT,3]

    float* out      = (float*)d_out;
    float* pPatches = out;                                    // B*NT*K*3
    float* pIdx     = pPatches + (size_t)BB * NT * KK * 3;    // B*NT*K*2
    float* pSize    = pIdx     + (size_t)BB * NT * KK * 2;    // B*NT
    float* pRad     = pSize    + (size_t)BB * NT;             // B
    float* pDist    = pRad     + BB;                          // B*NT*K

    dim3 grid(BB * (NT / TGT_PER_BLOCK));   // 128 blocks
    dim3 block(TGT_PER_BLOCK);              // 128 threads = 4 waves
    hipLaunchKernelGGL(group_points_knn_wmma, grid, block, 0, stream,
                       source, target, pPatches, pIdx, pSize, pRad, pDist);
}